// SpectralHypergraphConvLayer_816043786313
// MI455X (gfx1250) — compile-verified
//
#include <hip/hip_runtime.h>
#include <math.h>

// ---------------------------------------------------------------------------
// SpectralHypergraphConvLayer for MI455X (gfx1250), fp32 end-to-end.
//   K1: he_feat[hid]      += feat[pid] * w          (hw f32 atomic scatter)
//   K2: per-hyperedge attention -> he_weighted      (VALU, tiny)
//   K3: cluster_feat[pid] += he_weighted[hid] * w   (hw f32 atomic scatter)
//   K4: per-node MLP fused, V_WMMA_F32_16X16X4_F32, 32 rows/wave with
//       B-fragment register reuse across two 16-row sub-tiles.
// ---------------------------------------------------------------------------

#define N_P      50000
#define N_HE     5000
#define N_E      800000
#define D_IN     128
#define D_HID    256
#define D_OUT    128
#define N_HEADS  4
#define HEAD_DIM 64
#define MROWS    32   // rows per wave in the MLP kernel (2 x 16 sub-tiles)

typedef float v2f __attribute__((ext_vector_type(2)));
typedef float v8f __attribute__((ext_vector_type(8)));

__device__ __forceinline__ v8f zero8() {
  v8f z;
  for (int i = 0; i < 8; ++i) z[i] = 0.f;
  return z;
}

__device__ __forceinline__ v8f wmma4(v2f a, v2f b, v8f c) {
  return __builtin_amdgcn_wmma_f32_16x16x4_f32(false, a, false, b, (short)0, c,
                                               false, false);
}

// Two 16x16 fp32 C tiles sharing one registered B panel:
//   C0 += A0(16xK) * B(:, n0:n0+16) ; C1 += A1(16xK) * B(:, n0:n0+16)
// a0/a1 point at A row (lo), column (2*hi) of each sub-tile (float2 loads).
// B fragment: lane(lo,hi) holds rows (k+2*hi, k+2*hi+1) at column n0+lo.
template <int K>
__device__ __forceinline__ void gemm2(const float* __restrict__ a0,
                                      const float* __restrict__ a1,
                                      const float* __restrict__ B, int ldb,
                                      int n0, int lo, int hi,
                                      v8f& c0, v8f& c1) {
  v2f breg[K / 4];
#pragma unroll
  for (int k = 0; k < K; k += 4) {
    breg[k / 4].x = B[(k + 2 * hi) * ldb + n0 + lo];
    breg[k / 4].y = B[(k + 2 * hi + 1) * ldb + n0 + lo];
  }
#pragma unroll
  for (int k = 0; k < K; k += 4) {
    v2f a = *(const v2f*)(a0 + k);
    c0 = wmma4(a, breg[k / 4], c0);
  }
#pragma unroll
  for (int k = 0; k < K; k += 4) {
    v2f a = *(const v2f*)(a1 + k);
    c1 = wmma4(a, breg[k / 4], c1);
  }
}

// ---------------------------------------------------------------------------
// K1 / K3: edge scatter with hardware float atomics. One thread per
// (edge, 4-float chunk); edge index is wave-uniform -> SMEM metadata loads.
// ---------------------------------------------------------------------------
__global__ __launch_bounds__(256) void scatter_edges_kernel(
    const float* __restrict__ src,      // gathered by src_idx
    const float* __restrict__ edge_w,
    const int* __restrict__ src_idx,
    const int* __restrict__ dst_idx,
    float* __restrict__ dst)            // scattered by dst_idx (atomic +=)
{
  long tid = (long)blockIdx.x * blockDim.x + threadIdx.x;
  if (tid >= (long)N_E * 32) return;
  // All 32 lanes of a wave share one edge: force wave-uniform (SGPR) index.
  int e = __builtin_amdgcn_readfirstlane((int)(tid >> 5));
  int c = (int)(tid & 31);
  int si = src_idx[e];
  int di = dst_idx[e];
  float w = edge_w[e];
  const float4 v = *(const float4*)(src + (long)si * D_IN + c * 4);
  float* d = dst + (long)di * D_IN + c * 4;
  unsafeAtomicAdd(d + 0, v.x * w);
  unsafeAtomicAdd(d + 1, v.y * w);
  unsafeAtomicAdd(d + 2, v.z * w);
  unsafeAtomicAdd(d + 3, v.w * w);
}

// ---------------------------------------------------------------------------
// K2: hyperedge attention. One block (256 threads = N_HEADS*HEAD_DIM) per
// hyperedge. h = relu(he_feat @ Wh1 + bh1); attn = sigmoid(h . Wh2 + bh2);
// he_attn = sum_h attn*Wfuse; he_weighted = he_feat * he_attn.
// ---------------------------------------------------------------------------
__global__ __launch_bounds__(256) void attention_kernel(
    const float* __restrict__ he_feat,
    const float* __restrict__ Wh1, const float* __restrict__ bh1,
    const float* __restrict__ Wh2, const float* __restrict__ bh2,
    const float* __restrict__ Wfuse,
    float* __restrict__ he_weighted)
{
  __shared__ float row[D_IN];
  __shared__ float prod[N_HEADS * HEAD_DIM];
  __shared__ float att[N_HEADS];
  __shared__ float he_attn;

  const int he = blockIdx.x;
  const int t = threadIdx.x;

  if (t < D_IN) row[t] = he_feat[(long)he * D_IN + t];
  __syncthreads();

  const int hd = t >> 6;   // / HEAD_DIM
  const int k  = t & 63;   // % HEAD_DIM
  float acc = bh1[hd * HEAD_DIM + k];
  const float* w = Wh1 + (long)hd * D_IN * HEAD_DIM + k;
  for (int d = 0; d < D_IN; ++d) acc += row[d] * w[d * HEAD_DIM];
  float h = fmaxf(acc, 0.f);
  prod[t] = h * Wh2[hd * HEAD_DIM + k];
  __syncthreads();

  if (t < N_HEADS) {
    float s = 0.f;
    for (int i = 0; i < HEAD_DIM; ++i) s += prod[t * HEAD_DIM + i];
    att[t] = 1.f / (1.f + __expf(-(s + bh2[t])));
  }
  __syncthreads();
  if (t == 0) {
    float s = 0.f;
    for (int i = 0; i < N_HEADS; ++i) s += att[i] * Wfuse[i];
    he_attn = s;
  }
  __syncthreads();
  if (t < D_IN) he_weighted[(long)he * D_IN + t] = row[t] * he_attn;
}

// ---------------------------------------------------------------------------
// K4: fused per-node MLP on fp32 WMMA. One wave handles MROWS=32 nodes as
// two 16-row sub-tiles that share each B panel (register reuse).
//   self_f = feat @ Wself + bself            -> cat_lds[:, :128]
//   clu_f  = cluster_feat @ Wclu + bclu      -> cat_lds[:, 128:]
//   h1 = relu(cat @ Wf1 + bf1)               -> h1_lds
//   h2 = relu(h1 @ Wf2 + bf2)                -> consumed on the fly:
//   z  = h2 @ Wf3 + bf3 ; fw = softmax(z)    (shfl_xor row reduction)
//   out = relu(self_f*fw0 + clu_f*fw1 + feat)
// ---------------------------------------------------------------------------
__global__ __launch_bounds__(32) void node_mlp_kernel(
    const float* __restrict__ feat,
    const float* __restrict__ cluster_feat,
    const float* __restrict__ Wself, const float* __restrict__ bself,
    const float* __restrict__ Wclu,  const float* __restrict__ bclu,
    const float* __restrict__ Wf1,   const float* __restrict__ bf1,
    const float* __restrict__ Wf2,   const float* __restrict__ bf2,
    const float* __restrict__ Wf3,   const float* __restrict__ bf3,
    float* __restrict__ out)
{
  __shared__ float cat_lds[MROWS * (2 * D_OUT)];  // 32 x 256 (self_f | clu_f)
  __shared__ float h1_lds[MROWS * D_HID];         // 32 x 256

  const int lane = threadIdx.x;
  const int lo = lane & 15;
  const int hi = lane >> 4;
  const int m0 = blockIdx.x * MROWS;

  // Clamp A-row indices so every lane reads valid memory (EXEC must stay all
  // ones for WMMA); out-of-range results are discarded at the guarded store.
  const int ra0 = min(m0 + lo, N_P - 1);
  const int ra1 = min(m0 + 16 + lo, N_P - 1);

  // ---- self_f -> cat[:, 0:128] ----
  {
    const float* a0 = feat + (long)ra0 * D_IN + 2 * hi;
    const float* a1 = feat + (long)ra1 * D_IN + 2 * hi;
    for (int n0 = 0; n0 < D_OUT; n0 += 16) {
      v8f c0 = zero8(), c1 = zero8();
      gemm2<D_IN>(a0, a1, Wself, D_OUT, n0, lo, hi, c0, c1);
      float bb = bself[n0 + lo];
      for (int r = 0; r < 8; ++r) {
        cat_lds[(r + 8 * hi) * (2 * D_OUT) + n0 + lo] = c0[r] + bb;
        cat_lds[(16 + r + 8 * hi) * (2 * D_OUT) + n0 + lo] = c1[r] + bb;
      }
    }
  }
  // ---- clu_f -> cat[:, 128:256] ----
  {
    const float* a0 = cluster_feat + (long)ra0 * D_IN + 2 * hi;
    const float* a1 = cluster_feat + (long)ra1 * D_IN + 2 * hi;
    for (int n0 = 0; n0 < D_OUT; n0 += 16) {
      v8f c0 = zero8(), c1 = zero8();
      gemm2<D_IN>(a0, a1, Wclu, D_OUT, n0, lo, hi, c0, c1);
      float bb = bclu[n0 + lo];
      for (int r = 0; r < 8; ++r) {
        cat_lds[(r + 8 * hi) * (2 * D_OUT) + D_OUT + n0 + lo] = c0[r] + bb;
        cat_lds[(16 + r + 8 * hi) * (2 * D_OUT) + D_OUT + n0 + lo] = c1[r] + bb;
      }
    }
  }
  __syncthreads();

  // ---- h1 = relu(cat @ Wf1 + bf1) -> h1_lds ----
  {
    const float* a0 = cat_lds + lo * (2 * D_OUT) + 2 * hi;
    const float* a1 = cat_lds + (16 + lo) * (2 * D_OUT) + 2 * hi;
    for (int n0 = 0; n0 < D_HID; n0 += 16) {
      v8f c0 = zero8(), c1 = zero8();
      gemm2<2 * D_OUT>(a0, a1, Wf1, D_HID, n0, lo, hi, c0, c1);
      float bb = bf1[n0 + lo];
      for (int r = 0; r < 8; ++r) {
        h1_lds[(r + 8 * hi) * D_HID + n0 + lo] = fmaxf(c0[r] + bb, 0.f);
        h1_lds[(16 + r + 8 * hi) * D_HID + n0 + lo] = fmaxf(c1[r] + bb, 0.f);
      }
    }
  }
  __syncthreads();

  // ---- h2 = relu(h1 @ Wf2 + bf2), folded directly into z = h2 @ Wf3 ----
  // C layout: tile element (m = s*16 + r + 8*hi, n = t*16 + lo).
  float p0[2][8], p1[2][8];
  for (int s = 0; s < 2; ++s)
    for (int r = 0; r < 8; ++r) { p0[s][r] = 0.f; p1[s][r] = 0.f; }
  {
    const float* a0 = h1_lds + lo * D_HID + 2 * hi;
    const float* a1 = h1_lds + (16 + lo) * D_HID + 2 * hi;
    for (int t = 0; t < 8; ++t) {
      v8f c0 = zero8(), c1 = zero8();
      gemm2<D_HID>(a0, a1, Wf2, D_HID / 2, t * 16, lo, hi, c0, c1);
      float bb = bf2[t * 16 + lo];
      int n = t * 16 + lo;
      float w3a = Wf3[2 * n + 0];
      float w3b = Wf3[2 * n + 1];
      for (int r = 0; r < 8; ++r) {
        float h0 = fmaxf(c0[r] + bb, 0.f);
        float h1v = fmaxf(c1[r] + bb, 0.f);
        p0[0][r] += h0 * w3a;  p1[0][r] += h0 * w3b;
        p0[1][r] += h1v * w3a; p1[1][r] += h1v * w3b;
      }
    }
  }

  // ---- fw = softmax(z, axis=-1) via half-wave row reductions ----
  float fw0[2][8], fw1[2][8];
  for (int s = 0; s < 2; ++s) {
    for (int r = 0; r < 8; ++r) {
      float a = p0[s][r], b = p1[s][r];
      for (int sh = 1; sh < 16; sh <<= 1) {
        a += __shfl_xor(a, sh, 32);
        b += __shfl_xor(b, sh, 32);
      }
      float z0 = a + bf3[0];
      float z1 = b + bf3[1];
      float mx = fmaxf(z0, z1);
      float e0 = __expf(z0 - mx);
      float e1 = __expf(z1 - mx);
      float inv = 1.f / (e0 + e1);
      fw0[s][r] = e0 * inv;
      fw1[s][r] = e1 * inv;
    }
  }

  // ---- out = relu(self_f*fw0 + clu_f*fw1 + feat), guarded store ----
  for (int s = 0; s < 2; ++s) {
    for (int r = 0; r < 8; ++r) {
      int m = s * 16 + r + 8 * hi;
      int grow = m0 + m;
      if (grow < N_P) {
        for (int nn = lo; nn < D_IN; nn += 16) {
          float sf = cat_lds[m * (2 * D_OUT) + nn];
          float cf = cat_lds[m * (2 * D_OUT) + D_OUT + nn];
          float fused = sf * fw0[s][r] + cf * fw1[s][r];
          float o = fused + feat[(long)grow * D_IN + nn];
          out[(long)grow * D_IN + nn] = fmaxf(o, 0.f);
        }
      }
    }
  }
}

// ---------------------------------------------------------------------------
extern "C" void kernel_launch(void* const* d_in, const int* in_sizes, int n_in,
                              void* d_out, int out_size, void* d_ws,
                              size_t ws_size, hipStream_t stream) {
  const float* feat   = (const float*)d_in[0];
  const float* edge_w = (const float*)d_in[1];
  const float* Wself  = (const float*)d_in[2];
  const float* bself  = (const float*)d_in[3];
  const float* Wclu   = (const float*)d_in[4];
  const float* bclu   = (const float*)d_in[5];
  const float* Wh1    = (const float*)d_in[6];
  const float* bh1    = (const float*)d_in[7];
  const float* Wh2    = (const float*)d_in[8];
  const float* bh2    = (const float*)d_in[9];
  const float* Wfuse  = (const float*)d_in[10];
  const float* Wf1    = (const float*)d_in[11];
  const float* bf1    = (const float*)d_in[12];
  const float* Wf2    = (const float*)d_in[13];
  const float* bf2    = (const float*)d_in[14];
  const float* Wf3    = (const float*)d_in[15];
  const float* bf3    = (const float*)d_in[16];
  const int* edge_pid = (const int*)d_in[17];
  const int* edge_hid = (const int*)d_in[18];

  // Workspace layout (floats): he_feat | he_weighted | cluster_feat
  float* he_feat      = (float*)d_ws;
  float* he_weighted  = he_feat + (size_t)N_HE * D_IN;
  float* cluster_feat = he_weighted + (size_t)N_HE * D_IN;

  hipMemsetAsync(he_feat, 0, (size_t)N_HE * D_IN * sizeof(float), stream);
  hipMemsetAsync(cluster_feat, 0, (size_t)N_P * D_IN * sizeof(float), stream);

  const long nscatter = (long)N_E * 32;
  const int sblocks = (int)((nscatter + 255) / 256);

  // he_feat[hid] += feat[pid] * w
  scatter_edges_kernel<<<sblocks, 256, 0, stream>>>(feat, edge_w, edge_pid,
                                                    edge_hid, he_feat);
  // attention -> he_weighted
  attention_kernel<<<N_HE, 256, 0, stream>>>(he_feat, Wh1, bh1, Wh2, bh2,
                                             Wfuse, he_weighted);
  // cluster_feat[pid] += he_weighted[hid] * w
  scatter_edges_kernel<<<sblocks, 256, 0, stream>>>(he_weighted, edge_w,
                                                    edge_hid, edge_pid,
                                                    cluster_feat);
  // fused node MLP (WMMA fp32, 32 rows/wave)
  node_mlp_kernel<<<(N_P + MROWS - 1) / MROWS, 32, 0, stream>>>(
      feat, cluster_feat, Wself, bself, Wclu, bclu, Wf1, bf1, Wf2, bf2, Wf3,
      bf3, (float*)d_out);
}